// Kernel_78761110274634
// MI455X (gfx1250) — compile-verified
//
#include <hip/hip_runtime.h>

#define BATCH 4096
#define NPATH 4864
#define YDIM  25
#define MPAD  28     // m padded to multiple of 4 (WMMA K)
#define TDIM  259
#define TPAD  272    // t padded to multiple of 16 (WMMA M)
#define ODIM  144

typedef float v2f __attribute__((ext_vector_type(2)));
typedef float v4f __attribute__((ext_vector_type(4)));
typedef float v8f __attribute__((ext_vector_type(8)));

#if __has_builtin(__builtin_amdgcn_global_load_async_to_lds_b128) && \
    __has_builtin(__builtin_amdgcn_s_wait_asynccnt)
#define USE_ASYNC_LDS 1
// builtin expects int4-vector pointers: (AS1 int4*, AS3 int4*, imm, imm)
typedef int i4v __attribute__((vector_size(16)));
typedef __attribute__((address_space(1))) i4v gi4;
typedef __attribute__((address_space(3))) i4v si4;
#else
#define USE_ASYNC_LDS 0
#endif

// ---------------- constants ----------------
__constant__ double c_fact[13] = {1.,1.,2.,6.,24.,120.,720.,5040.,40320.,
                                  362880.,3628800.,39916800.,479001600.};
// pairs p = bo*3+bi, bo/bi in {0,1,2} with l = index
__constant__ int c_nlf[9] = {1,1,1,1,3,3,1,3,5};
__constant__ int c_tb[9]  = {0,1,4,9,12,39,84,89,134};

// ---------------- Wigner 3j (double) ----------------
__device__ inline double wig3j(int j1,int j2,int j3,int m1,int m2,int m3){
  if (m1+m2+m3 != 0) return 0.0;
  if (j3 < abs(j1-j2) || j3 > j1+j2) return 0.0;
  if (abs(m1)>j1 || abs(m2)>j2 || abs(m3)>j3) return 0.0;
  double pre = sqrt(c_fact[j1+j2-j3]*c_fact[j1-j2+j3]*c_fact[-j1+j2+j3]/c_fact[j1+j2+j3+1]);
  pre *= sqrt(c_fact[j1+m1]*c_fact[j1-m1]*c_fact[j2+m2]*c_fact[j2-m2]*c_fact[j3+m3]*c_fact[j3-m3]);
  int t0 = 0; if (j2-j3-m1 > t0) t0 = j2-j3-m1; if (j1-j3+m2 > t0) t0 = j1-j3+m2;
  int t1 = j1+j2-j3; if (j1-m1 < t1) t1 = j1-m1; if (j2+m2 < t1) t1 = j2+m2;
  double s = 0.0;
  for (int t=t0; t<=t1; ++t){
    double d = c_fact[t]*c_fact[j3-j2+m1+t]*c_fact[j3-j1-m2+t]*
               c_fact[j1+j2-j3-t]*c_fact[j1-m1-t]*c_fact[j2+m2-t];
    s += ((t&1)? -1.0:1.0)/d;
  }
  return (((j1-j2-m3)&1)? -1.0:1.0)*pre*s;
}

// conj(A[a,i]) of the real->complex change-of-basis matrix for degree l
__device__ inline void Aconj(int l,int a,int i,double& re,double& im){
  const double s2 = 0.70710678118654752440;
  int M = a - l, m = i - l;
  re = 0.0; im = 0.0;
  if (M == 0){ if (m == 0) re = 1.0; }
  else if (M > 0){
    double sgn = (M & 1) ? -1.0 : 1.0;
    if (m ==  M) re =  sgn*s2;
    else if (m == -M) im = -sgn*s2;     // conj(i*(-1)^M/sqrt2)
  } else {
    if (m == -M) re =  s2;
    else if (m ==  M) im =  s2;         // conj(-i/sqrt2)
  }
}

// ---------------- kernel 1: build Cflat[TPAD][MPAD] ----------------
__global__ void cg_setup(float* __restrict__ Cflat){
  int c = blockIdx.x;                       // 0..18, one (pair,k) combo
  int p = 0, k = c;
  while (k >= c_nlf[p]) { k -= c_nlf[p]; ++p; }
  int l1 = p/3, l2 = p%3;
  int l3 = abs(l1-l2) + k;
  int I = 2*l1+1, J = 2*l2+1, M3 = 2*l3+1;

  __shared__ double Tt[225];                // 5*5*9 max
  __shared__ double CRe[225], CIm[225];
  __shared__ double red[2];

  int tid = threadIdx.x, NTh = blockDim.x;
  int nT = I*J*M3;
  for (int e = tid; e < nT; e += NTh){
    int a = e/(J*M3), r = e - a*J*M3;
    int b = r/M3,  cc = r - b*M3;
    Tt[e] = wig3j(l1,l2,l3, a-l1, b-l2, cc-l3);
  }
  __syncthreads();

  int nE = I*J*M3;
  for (int e = tid; e < nE; e += NTh){
    int i = e/(J*M3), r = e - i*J*M3;
    int j = r/M3,   mp = r - j*M3;
    double sre = 0.0, sim = 0.0;
    for (int a=0;a<I;++a){
      double a1r,a1i; Aconj(l1,a,i,a1r,a1i);
      if (a1r==0.0 && a1i==0.0) continue;
      for (int b=0;b<J;++b){
        double a2r,a2i; Aconj(l2,b,j,a2r,a2i);
        if (a2r==0.0 && a2i==0.0) continue;
        for (int cc=0;cc<M3;++cc){
          double t = Tt[(a*J+b)*M3+cc];
          if (t == 0.0) continue;
          double a3r,a3i; Aconj(l3,cc,mp,a3r,a3i);
          double pr = a1r*a2r - a1i*a2i, pi = a1r*a2i + a1i*a2r;
          double qr = pr*a3r - pi*a3i,  qi = pr*a3i + pi*a3r;
          sre += t*qr; sim += t*qi;
        }
      }
    }
    CRe[e] = sre; CIm[e] = sim;
  }
  __syncthreads();
  if (tid == 0){
    double nr=0.0, ni=0.0;
    for (int e=0;e<nE;++e){ nr += CRe[e]*CRe[e]; ni += CIm[e]*CIm[e]; }
    red[0]=nr; red[1]=ni;
  }
  __syncthreads();
  bool useRe = red[0] >= red[1];
  double nn = sqrt(useRe ? red[0] : red[1]);
  double scale = (nn > 0.0) ? 1.0/nn : 0.0;

  int yoff = l3*l3;                         // Y_OFF[lf] = lf^2
  int TB = c_tb[p];
  for (int e = tid; e < I*J; e += NTh){
    int i = e/J, j = e - i*J;
    int t = TB + (k*I + i)*J + j;
    for (int m=0;m<MPAD;++m){
      float val = 0.f;
      if (m >= yoff && m < yoff + M3){
        double v = useRe ? CRe[(i*J+j)*M3 + (m-yoff)] : CIm[(i*J+j)*M3 + (m-yoff)];
        val = (float)(v*scale);
      }
      Cflat[t*MPAD + m] = val;
    }
  }
  if (c == 0){                              // zero padding rows
    for (int e = tid; e < (TPAD-TDIM)*MPAD; e += NTh)
      Cflat[TDIM*MPAD + e] = 0.f;
  }
}

// ---------------- kernel 2: zero-padded Y ----------------
__global__ void pad_y(const float* __restrict__ Y, float* __restrict__ Ypad){
  int idx = blockIdx.x*256 + threadIdx.x;   // MPAD*BATCH exactly
  int m = idx >> 12;                        // /4096
  Ypad[idx] = (m < YDIM) ? Y[idx] : 0.f;
}

// ---------------- kernel 3: cyT = Cflat x Ypad via WMMA f32 16x16x4 ----------------
__global__ void cy_gemm(const float* __restrict__ Cflat,
                        const float* __restrict__ Ypad,
                        float* __restrict__ cyT){
  int lane = threadIdx.x;                   // one full wave32 per tile
  int half = lane >> 4, lm = lane & 15;
  int trow = blockIdx.y * 16;               // T tile
  int zcol = blockIdx.x * 16;               // batch tile
  v8f acc = {};
  #pragma unroll
  for (int kk = 0; kk < MPAD/4; ++kk){
    int k0 = kk*4 + 2*half;
    v2f a, b;
    // A (16x4 f32): lane = M, vgpr r holds K = r + 2*(lane/16)
    a.x = Cflat[(trow + lm)*MPAD + k0 + 0];
    a.y = Cflat[(trow + lm)*MPAD + k0 + 1];
    // B (4x16 f32): lane%16 = N, vgpr r holds K = r + 2*(lane/16)
    b.x = Ypad[(size_t)(k0 + 0)*BATCH + zcol + lm];
    b.y = Ypad[(size_t)(k0 + 1)*BATCH + zcol + lm];
    acc = __builtin_amdgcn_wmma_f32_16x16x4_f32(
        /*neg_a=*/false, a, /*neg_b=*/false, b,
        /*c_mod=*/(short)0, acc, /*reuse_a=*/false, /*reuse_b=*/false);
  }
  // D (16x16 f32): vgpr r -> M = r + 8*(lane/16), lane%16 = N
  #pragma unroll
  for (int r = 0; r < 8; ++r)
    cyT[(size_t)(trow + r + 8*half)*BATCH + zcol + lm] = acc[r];
}

// ---------------- kernel 4: main tensor-product expansion ----------------
template<int I,int J,int NLF,int ROFF,int TB,int R0,int C0>
__device__ inline void block_pair(const float* __restrict__ Rs,
                                  const float* __restrict__ cys,
                                  float nrm, float* __restrict__ outz, int tid){
  constexpr int COLS = 16*J, ROWS = 16*I;
  constexpr int CG = COLS/4;                // groups of 4 cols (16/48/80 all %4==0)
  for (int idx = tid; idx < ROWS*CG; idx += 256){
    int row = idx / CG;
    int c0  = (idx - row*CG)*4;
    int u = row / I, i = row - u*I;
    v4f res;
    #pragma unroll
    for (int q = 0; q < 4; ++q){
      int col = c0 + q;
      int v = col / J, j = col - v*J;
      float acc = 0.f;
      #pragma unroll
      for (int k = 0; k < NLF; ++k)
        acc += cys[TB + (k*I + i)*J + j] * Rs[ROFF + (u*16 + v)*NLF + k];
      res[q] = nrm * acc;
    }
    // streaming output: write-once, never read -> non-temporal b128
    __builtin_nontemporal_store(res,
        (v4f*)&outz[(size_t)(R0 + row)*ODIM + C0 + c0]);
  }
}

__global__ void tp_main(const float* __restrict__ R,
                        const float* __restrict__ cyT,
                        const float* __restrict__ norm,
                        float* __restrict__ out){
  int z = blockIdx.x;
  __shared__ float Rs[NPATH];
  __shared__ float cys[TDIM];
  __shared__ float ns[9];
  int tid = threadIdx.x;

#if USE_ASYNC_LDS
  {
    gi4* gsrc = (gi4*)(uintptr_t)(R + (size_t)z*NPATH);
    si4* ldst = (si4*)(uint32_t)(uintptr_t)(&Rs[0]);
    for (int i4 = tid; i4 < NPATH/4; i4 += 256)     // one b128 per issue
      __builtin_amdgcn_global_load_async_to_lds_b128(gsrc + i4, ldst + i4, 0, 0);
  }
#else
  {
    const v4f* Rv = (const v4f*)(R + (size_t)z*NPATH);
    v4f* Rsv = (v4f*)Rs;
    for (int i = tid; i < NPATH/4; i += 256) Rsv[i] = Rv[i];
  }
#endif
  for (int idx = tid; idx < TDIM; idx += 256) cys[idx] = cyT[(size_t)idx*BATCH + z];
  if (tid < 9) ns[tid] = norm[(size_t)tid*BATCH + z];
#if USE_ASYNC_LDS
  __builtin_amdgcn_s_wait_asynccnt(0);
#endif
  __syncthreads();

  float* outz = out + (size_t)z*ODIM*ODIM;
  //            I J NLF ROFF  TB  R0 C0
  block_pair<1,1,1,   0,   0,  0,  0>(Rs,cys,ns[0],outz,tid);
  block_pair<1,3,1, 256,   1,  0, 16>(Rs,cys,ns[1],outz,tid);
  block_pair<1,5,1, 512,   4,  0, 64>(Rs,cys,ns[2],outz,tid);
  block_pair<3,1,1, 768,   9, 16,  0>(Rs,cys,ns[3],outz,tid);
  block_pair<3,3,3,1024,  12, 16, 16>(Rs,cys,ns[4],outz,tid);
  block_pair<3,5,3,1792,  39, 16, 64>(Rs,cys,ns[5],outz,tid);
  block_pair<5,1,1,2560,  84, 64,  0>(Rs,cys,ns[6],outz,tid);
  block_pair<5,3,3,2816,  89, 64, 16>(Rs,cys,ns[7],outz,tid);
  block_pair<5,5,5,3584, 134, 64, 64>(Rs,cys,ns[8],outz,tid);
}

// ---------------- launch ----------------
extern "C" void kernel_launch(void* const* d_in, const int* in_sizes, int n_in,
                              void* d_out, int out_size, void* d_ws, size_t ws_size,
                              hipStream_t stream){
  const float* Y    = (const float*)d_in[0];   // [25, 4096]
  const float* R    = (const float*)d_in[1];   // [4096, 4864]
  const float* norm = (const float*)d_in[2];   // [3, 3, 4096]
  float* out = (float*)d_out;                  // [4096, 144, 144]

  float* Cflat = (float*)d_ws;                         // TPAD*MPAD
  float* Ypad  = Cflat + TPAD*MPAD;                    // MPAD*BATCH
  float* cyT   = Ypad + (size_t)MPAD*BATCH;            // TPAD*BATCH

  cg_setup<<<19, 64, 0, stream>>>(Cflat);
  pad_y<<<(MPAD*BATCH)/256, 256, 0, stream>>>(Y, Ypad);
  cy_gemm<<<dim3(BATCH/16, TPAD/16), 32, 0, stream>>>(Cflat, Ypad, cyT);
  tp_main<<<BATCH, 256, 0, stream>>>(R, cyT, norm, out);
}